// DensityLoss_1013612282417
// MI455X (gfx1250) — compile-verified
//
#include <hip/hip_runtime.h>
#include <hip/hip_bf16.h>
#include <math.h>
#include <stdint.h>

typedef __attribute__((ext_vector_type(2))) float v2f;
typedef __attribute__((ext_vector_type(8))) float v8f;

#define NPTS      4096
#define NSAMPLE   9
#define RADIUS_F  0.1f
#define R2        (RADIUS_F * RADIUS_F)
#define H_F       0.12f
#define INV_H2    (1.0f / (H_F * H_F))
#define EPS_F     1e-12f

#define THREADS             256
#define WAVES               (THREADS / 32)
#define ROWS_PER_WAVE       32                        // 2 WMMA row-tiles per wave
#define ROWS_PER_BLOCK      (WAVES * ROWS_PER_WAVE)   // 256
#define ROW_TILES_PER_BATCH (NPTS / ROWS_PER_BLOCK)   // 16
#define COL_TILES           (NPTS / 16)               // 256

#define USE_ASYNC_STAGE 1

// LDS layout (floats): pts[12288] | sq[4096] | bp[8192] | tiles[WAVES*512]
#define OFF_PTS   0
#define OFF_SQ    (NPTS * 3)
#define OFF_BP    (OFF_SQ + NPTS)
#define OFF_TILES (OFF_BP + NPTS * 2)
#define SMEM_FLOATS (OFF_TILES + WAVES * 512)         // 28672 floats = 112 KB

// Branchless insert into 5 ascending sorted registers (drop largest).
__device__ __forceinline__ void insert5(float x, float& m0, float& m1,
                                        float& m2, float& m3, float& m4) {
  float t;
  t = fminf(x, m0); x = fmaxf(x, m0); m0 = t;
  t = fminf(x, m1); x = fmaxf(x, m1); m1 = t;
  t = fminf(x, m2); x = fmaxf(x, m2); m2 = t;
  t = fminf(x, m3); x = fmaxf(x, m3); m3 = t;
  m4 = fminf(x, m4);
}

__global__ __launch_bounds__(THREADS)
void density_loss_kernel(const float* __restrict__ pred,
                         float* __restrict__ out, float scale) {
  extern __shared__ __align__(16) float smem[];
  float* pts   = smem + OFF_PTS;    // [NPTS*3] raw points
  float* sq    = smem + OFF_SQ;     // [NPTS]   |p|^2
  v2f*   bp    = (v2f*)(smem + OFF_BP);   // [COL_TILES*32] WMMA-ready operands
  float* tiles = smem + OFF_TILES;  // [WAVES][512] per-wave dot-product tiles

  const int tid      = threadIdx.x;
  const int b        = blockIdx.x / ROW_TILES_PER_BATCH;
  const int rowBlock = blockIdx.x % ROW_TILES_PER_BATCH;
  const int wid      = tid >> 5;
  const int lane     = tid & 31;
  const int hlane    = lane & 15;
  const bool lowHalf = lane < 16;

  // ---- Stage this batch's points into LDS ----
#if USE_ASYNC_STAGE
  {
    // gfx1250 async copy: global -> LDS without a VGPR round trip (ASYNCcnt).
    const uint64_t gbase = (uint64_t)(uintptr_t)(pred + (size_t)b * NPTS * 3);
    const uint32_t lbase = (uint32_t)(uintptr_t)pts;  // flat-LDS low bits == LDS addr
    for (int i = tid; i < (NPTS * 3) / 4; i += THREADS) {
      uint64_t ga = gbase + (uint64_t)i * 16u;
      uint32_t la = lbase + (uint32_t)i * 16u;
      asm volatile("global_load_async_to_lds_b128 %0, %1, off"
                   :: "v"(la), "v"(ga) : "memory");
    }
    asm volatile("s_wait_asynccnt 0" ::: "memory");
  }
#else
  {
    const float4* gp4 = (const float4*)(pred + (size_t)b * NPTS * 3);
    float4* p4 = (float4*)pts;
    for (int i = tid; i < (NPTS * 3) / 4; i += THREADS) p4[i] = gp4[i];
  }
#endif
  __syncthreads();

  // ---- Pack squared norms + WMMA-ready A/B operand pairs ----
  // f32 16x4 A / 4x16 B layout: lanes 0-15 carry {x,y} (K=0,1) of point
  // (tile*16 + lane), lanes 16-31 carry {z,0} (K=2,3).
  for (int i = tid; i < NPTS; i += THREADS) {
    float x = pts[i * 3], y = pts[i * 3 + 1], z = pts[i * 3 + 2];
    sq[i] = x * x + y * y + z * z;
    int t = i >> 4, n = i & 15;
    v2f lo; lo[0] = x; lo[1] = y;
    v2f hi; hi[0] = z; hi[1] = 0.0f;
    bp[t * 32 + n]      = lo;
    bp[t * 32 + 16 + n] = hi;
  }
  __syncthreads();

  // ---- This wave's 32 query rows = two 16-row WMMA tiles ----
  const int rowBase = rowBlock * ROWS_PER_BLOCK + wid * ROWS_PER_WAVE;
  const int rt0 = rowBase >> 4;           // first row-tile index
  const v2f av0 = bp[rt0 * 32 + lane];
  const v2f av1 = bp[(rt0 + 1) * 32 + lane];
  const float sqRow = sq[rowBase + lane]; // each lane owns row rowBase+lane

  int   cnt   = 0;
  float first = 0.0f;
  float m0 = 3.0e38f, m1 = 3.0e38f, m2 = 3.0e38f, m3 = 3.0e38f, m4 = 3.0e38f;

  float* tb = tiles + wid * 512;
  const int rShift = lowHalf ? 0 : 8;     // C/D layout: vgpr r -> row r (+8 hi half)
  const float* rowp = tb + lane * 16;     // lane's row: works for both tiles
                                          // (256 + (lane-16)*16 == lane*16)

  for (int t = 0; t < COL_TILES; ++t) {
    const int col0 = t * 16;
    const v2f bv = bp[t * 32 + lane];     // one clean b64 load, no predication

    v8f c0 = {0.f,0.f,0.f,0.f,0.f,0.f,0.f,0.f};
    v8f c1 = {0.f,0.f,0.f,0.f,0.f,0.f,0.f,0.f};
    c0 = __builtin_amdgcn_wmma_f32_16x16x4_f32(false, av0, false, bv,
                                               (short)0, c0, false, false);
    c1 = __builtin_amdgcn_wmma_f32_16x16x4_f32(false, av1, false, bv,
                                               (short)0, c1, false, false);

    // Spill both 16x16 dot tiles row-major (rows 0-15 then 16-31).
#pragma unroll
    for (int r = 0; r < 8; ++r) {
      tb[(rShift + r) * 16 + hlane]       = c0[r];
      tb[256 + (rShift + r) * 16 + hlane] = c1[r];
    }
    __builtin_amdgcn_wave_barrier();      // LDS is in-order within a wave

    // All 32 lanes scan their own row over these 16 columns (index order).
#pragma unroll
    for (int g = 0; g < 4; ++g) {
      float4 dg = ((const float4*)rowp)[g];
      float4 sg = ((const float4*)(sq + col0))[g];
      float d0 = fmaf(-2.f, dg.x, sqRow + sg.x);
      float d1 = fmaf(-2.f, dg.y, sqRow + sg.y);
      float d2 = fmaf(-2.f, dg.z, sqRow + sg.z);
      float d3 = fmaf(-2.f, dg.w, sqRow + sg.w);
      float mn = fminf(fminf(d0, d1), fminf(d2, d3));
      if (mn <= R2 && cnt < NSAMPLE) {    // rare: skip accept machinery otherwise
        float dv[4] = {d0, d1, d2, d3};
#pragma unroll
        for (int k = 0; k < 4; ++k) {
          float d = dv[k];
          if (d <= R2 && cnt < NSAMPLE) { // accept first 9 in index order
            if (cnt == 0) first = d;
            ++cnt;
            insert5(d, m0, m1, m2, m3, m4);
          }
        }
      }
    }
    __builtin_amdgcn_wave_barrier();
  }

  // Padding semantics: missing slots replicate the first hit (cnt >= 1).
  const int pad = NSAMPLE - cnt;
#pragma unroll
  for (int k = 0; k < NSAMPLE - 1; ++k)
    if (k < pad) insert5(first, m0, m1, m2, m3, m4);

  // Drop the global minimum (m0); ranks 2..5 are m1..m4.
  float vals[4] = {m1, m2, m3, m4};
  float localSum = 0.0f;
#pragma unroll
  for (int k = 0; k < 4; ++k) {
    float d = fmaxf(vals[k], EPS_F);
    localSum += RADIUS_F - __fsqrt_rn(d) * __expf(-d * INV_H2);
  }

  // Full-wave reduction, then one atomic per wave.
#pragma unroll
  for (int off = 16; off >= 1; off >>= 1)
    localSum += __shfl_xor(localSum, off, 32);
  if (lane == 0) atomicAdd(out, localSum * scale);
}

extern "C" void kernel_launch(void* const* d_in, const int* in_sizes, int n_in,
                              void* d_out, int out_size, void* d_ws, size_t ws_size,
                              hipStream_t stream) {
  const float* pred = (const float*)d_in[0];
  float* out = (float*)d_out;
  const int B = in_sizes[0] / (NPTS * 3);  // 8 for the reference shapes

  hipMemsetAsync(out, 0, sizeof(float) * (size_t)out_size, stream);

  const float scale = 1.0f / (float)(B * NPTS * 4);  // mean over [B,N,TOPK-1]
  dim3 grid(B * ROW_TILES_PER_BATCH);
  dim3 block(THREADS);
  size_t smemBytes = (size_t)SMEM_FLOATS * sizeof(float);  // 114,688 B < 320 KB
  density_loss_kernel<<<grid, block, smemBytes, stream>>>(pred, out, scale);
}